// WaveletTokenizer_1503238553779
// MI455X (gfx1250) — compile-verified
//
#include <hip/hip_runtime.h>
#include <math.h>

#define VOCAB   4096
#define NQ      65536      // B*L = 4*16384
#define DDIM    3
#define DECAYF  0.99f
#define OMDF    0.01f      // 1 - DECAY
#define EPSF    1e-5f
#define BETAF   0.25f

#define NTILE   (VOCAB / 16)   // 256 code tiles of 16

// workspace layout (float units)
#define OFF_SSE   0
#define OFF_N     1
#define OFF_CNT   2
#define OFF_WSUM  (2 + VOCAB)             // V x 3
#define OFF_E4    16388                   // V x float4, 16B aligned (16388*4 % 16 == 0)

// output layout (float units), reference tuple order
#define O_QUANT   0                       // 196608
#define O_IDX     196608                  // 65536 (stored as float(idx))
#define O_LOSS    262144                  // 1
#define O_EMB     262145                  // 12288
#define O_NCS     274433                  // 4096
#define O_NEMAW   278529                  // 12288

typedef __attribute__((ext_vector_type(2))) float v2f;
typedef __attribute__((ext_vector_type(8))) float v8f;

// ---------------- kernel 0: zero accumulators, build {e0,e1,e2,|e|^2} table ----
__global__ __launch_bounds__(256) void k_init(const float* __restrict__ embed,
                                              float* __restrict__ ws) {
  int v = blockIdx.x * blockDim.x + threadIdx.x;
  if (v < VOCAB) {
    float e0 = embed[3 * v + 0];
    float e1 = embed[3 * v + 1];
    float e2 = embed[3 * v + 2];
    float4* e4 = (float4*)(ws + OFF_E4);
    e4[v] = make_float4(e0, e1, e2, e0 * e0 + e1 * e1 + e2 * e2);
    ws[OFF_CNT + v] = 0.0f;
    ws[OFF_WSUM + 3 * v + 0] = 0.0f;
    ws[OFF_WSUM + 3 * v + 1] = 0.0f;
    ws[OFF_WSUM + 3 * v + 2] = 0.0f;
    if (v == 0) { ws[OFF_SSE] = 0.0f; ws[OFF_N] = 0.0f; }
  }
}

// ---------------- kernel 1: WMMA distance scan + argmin + scatter -------------
// One wave handles 16 queries. A(16x4) row m = (-2x0,-2x1,-2x2, 1),
// B(4x16) col n = (e0,e1,e2,|e|^2)  =>  D[m][n] = -2*x.e + |e|^2.
__global__ __launch_bounds__(256) void k_assign(const float* __restrict__ feats,
                                                float* __restrict__ out,
                                                float* __restrict__ ws) {
  const int lane  = threadIdx.x & 31;
  const int wave  = threadIdx.x >> 5;
  const int q0    = (blockIdx.x * 8 + wave) * 16;
  const int col   = lane & 15;   // M for A-load, N for D
  const int khalf = lane >> 4;   // 0: K={0,1}, 1: K={2,3}

  // Build A operand (2 VGPRs) for query row m = col
  const int qa = q0 + col;
  const float x0 = feats[3 * qa + 0];
  const float x1 = feats[3 * qa + 1];
  const float x2 = feats[3 * qa + 2];
  v2f a;
  a.x = khalf ? (-2.0f * x2) : (-2.0f * x0);
  a.y = khalf ? 1.0f         : (-2.0f * x1);

  const float4* __restrict__ e4 = (const float4*)(ws + OFF_E4);

  float best[8];
  int   bestn[8];
#pragma unroll
  for (int r = 0; r < 8; ++r) { best[r] = 3.402823466e38f; bestn[r] = 0; }

  v8f c0 = {};
  // 4 independent load->WMMA chains per trip; unconditional clamped prefetch.
  for (int t = 0; t < NTILE; t += 4) {
    int pf = t + 8; if (pf > NTILE - 4) pf = NTILE - 4;
    __builtin_prefetch(&e4[pf * 16 + col], 0, 3);
#pragma unroll
    for (int u = 0; u < 4; ++u) {
      const float4 e = e4[(t + u) * 16 + col];
      v2f b;
      b.x = khalf ? e.z : e.x;
      b.y = khalf ? e.w : e.y;
      // v_wmma_f32_16x16x4_f32 : D = A*B + 0
      v8f d = __builtin_amdgcn_wmma_f32_16x16x4_f32(
          /*neg_a=*/false, a, /*neg_b=*/false, b,
          /*c_mod=*/(short)0, c0, /*reuse_a=*/false, /*reuse_b=*/false);
      const int n = (t + u) * 16 + col;
#pragma unroll
      for (int r = 0; r < 8; ++r) {
        const bool lt = d[r] < best[r];
        best[r]  = lt ? d[r] : best[r];
        bestn[r] = lt ? n    : bestn[r];
      }
    }
  }

  // branchless argmin across the 16 lanes of each half
  // (xor masks < 16 keep lanes 0-15 / 16-31 separate on wave32)
#pragma unroll
  for (int r = 0; r < 8; ++r) {
#pragma unroll
    for (int off = 8; off >= 1; off >>= 1) {
      const float ob = __shfl_xor(best[r],  off, 32);
      const int   on = __shfl_xor(bestn[r], off, 32);
      const int take = (int)(ob < best[r]) |
                       ((int)(ob == best[r]) & (int)(on < bestn[r]));
      best[r]  = take ? ob : best[r];
      bestn[r] = take ? on : bestn[r];
    }
  }

  // writer lanes: lane 0 -> rows 0..7, lane 16 -> rows 8..15 (static indexing)
  float sq = 0.0f;
  if (col == 0) {
#pragma unroll
    for (int r = 0; r < 8; ++r) {
      const int m   = r + 8 * khalf;
      const int qq  = q0 + m;
      const int idx = bestn[r];
      out[O_IDX + qq] = (float)idx;
      const float4 e = e4[idx];
      const float f0 = feats[3 * qq + 0];
      const float f1 = feats[3 * qq + 1];
      const float f2 = feats[3 * qq + 2];
      const float d0 = e.x - f0, d1 = e.y - f1, d2 = e.z - f2;
      out[O_QUANT + 3 * qq + 0] = f0 + d0;   // straight-through quant
      out[O_QUANT + 3 * qq + 1] = f1 + d1;
      out[O_QUANT + 3 * qq + 2] = f2 + d2;
      sq += d0 * d0 + d1 * d1 + d2 * d2;
      atomicAdd(&ws[OFF_CNT + idx], 1.0f);
      atomicAdd(&ws[OFF_WSUM + 3 * idx + 0], f0);
      atomicAdd(&ws[OFF_WSUM + 3 * idx + 1], f1);
      atomicAdd(&ws[OFF_WSUM + 3 * idx + 2], f2);
    }
  }
  // only lanes 0 and 16 hold nonzero partials: one xor-16 shuffle combines them
  sq += __shfl_xor(sq, 16, 32);
  if (lane == 0) atomicAdd(&ws[OFF_SSE], sq);
}

// ---------------- kernel 2: EMA blend + global n reduction --------------------
__global__ __launch_bounds__(256) void k_ema(const float* __restrict__ ema_cs,
                                             const float* __restrict__ ema_w,
                                             float* __restrict__ out,
                                             float* __restrict__ ws) {
  int v = blockIdx.x * blockDim.x + threadIdx.x;
  if (v < VOCAB) {
    float ncs = DECAYF * ema_cs[v] + OMDF * ws[OFF_CNT + v];
    out[O_NCS + v] = ncs;
    atomicAdd(&ws[OFF_N], ncs);
    out[O_NEMAW + 3 * v + 0] = DECAYF * ema_w[3 * v + 0] + OMDF * ws[OFF_WSUM + 3 * v + 0];
    out[O_NEMAW + 3 * v + 1] = DECAYF * ema_w[3 * v + 1] + OMDF * ws[OFF_WSUM + 3 * v + 1];
    out[O_NEMAW + 3 * v + 2] = DECAYF * ema_w[3 * v + 2] + OMDF * ws[OFF_WSUM + 3 * v + 2];
  }
}

// ---------------- kernel 3: normalize embed, finalize loss --------------------
__global__ __launch_bounds__(256) void k_final(float* __restrict__ out,
                                               const float* __restrict__ ws) {
  int v = blockIdx.x * blockDim.x + threadIdx.x;
  if (v < VOCAB) {
    const float n   = ws[OFF_N];
    const float ncs = out[O_NCS + v];
    const float cs  = (ncs + EPSF) / (n + (float)VOCAB * EPSF) * n;
    const float w0  = out[O_NEMAW + 3 * v + 0] / cs;
    const float w1  = out[O_NEMAW + 3 * v + 1] / cs;
    const float w2  = out[O_NEMAW + 3 * v + 2] / cs;
    float norm = fmaxf(sqrtf(w1 * w1 + w2 * w2), EPSF);
    out[O_EMB + 3 * v + 0] = w0;
    out[O_EMB + 3 * v + 1] = w1 / norm;
    out[O_EMB + 3 * v + 2] = w2 / norm;
    if (v == 0) {
      out[O_LOSS] = (1.0f + BETAF) * ws[OFF_SSE] / (float)(NQ * DDIM);
    }
  }
}

extern "C" void kernel_launch(void* const* d_in, const int* in_sizes, int n_in,
                              void* d_out, int out_size, void* d_ws, size_t ws_size,
                              hipStream_t stream) {
  (void)in_sizes; (void)n_in; (void)out_size; (void)ws_size;
  const float* feats  = (const float*)d_in[0];
  const float* embed  = (const float*)d_in[1];
  const float* ema_cs = (const float*)d_in[2];
  const float* ema_w  = (const float*)d_in[3];
  float* out = (float*)d_out;
  float* ws  = (float*)d_ws;

  k_init  <<<VOCAB / 256, 256, 0, stream>>>(embed, ws);
  k_assign<<<NQ / (16 * 8), 256, 0, stream>>>(feats, out, ws);   // 512 blocks x 8 waves x 16 queries
  k_ema   <<<VOCAB / 256, 256, 0, stream>>>(ema_cs, ema_w, out, ws);
  k_final <<<VOCAB / 256, 256, 0, stream>>>(out, ws);
}